// LineSpectralPairsStabilityCheck_27032524161516
// MI455X (gfx1250) — compile-verified
//
#include <hip/hip_runtime.h>
#include <math.h>

#define NP1 33                  // LSP_ORDER + 1
#define ROWS_PER_BLOCK 256
#define TILE_WORDS (ROWS_PER_BLOCK * NP1)   // 8448 floats = 33792 bytes LDS
#define TILE_VEC4  (TILE_WORDS / 4)         // 2112 float4 chunks

#define K_PI   3.14159265358979323846
#define K_MIND (0.01 * K_PI / 33.0)
__device__ static constexpr float kMinD = (float)K_MIND;
__device__ static constexpr float kPiF  = (float)K_PI;

// 16-byte vector type matching the async builtins' expected pointee type.
typedef int v4i __attribute__((ext_vector_type(4)));
typedef __attribute__((address_space(1))) v4i* gv4i_p;
typedef __attribute__((address_space(3))) v4i* lv4i_p;
typedef __attribute__((address_space(1))) void* gvoid_p;
typedef __attribute__((address_space(3))) void* lvoid_p;

// ---------------- CDNA5 async global<->LDS helpers ----------------
__device__ static inline void async_load_b128(const float* gsrc, float* ldst) {
#if __has_builtin(__builtin_amdgcn_global_load_async_to_lds_b128)
    __builtin_amdgcn_global_load_async_to_lds_b128(
        (gv4i_p)(gvoid_p)gsrc,
        (lv4i_p)(lvoid_p)ldst,
        0, 0);
#else
    unsigned lo = (unsigned)(unsigned long long)(lvoid_p)ldst;
    asm volatile("global_load_async_to_lds_b128 %0, %1, off"
                 :: "v"(lo), "v"(gsrc) : "memory");
#endif
}

__device__ static inline void async_store_b128(float* gdst, const float* lsrc) {
#if __has_builtin(__builtin_amdgcn_global_store_async_from_lds_b128)
    __builtin_amdgcn_global_store_async_from_lds_b128(
        (gv4i_p)(gvoid_p)gdst,
        (lv4i_p)(lvoid_p)lsrc,
        0, 0);
#else
    unsigned lo = (unsigned)(unsigned long long)(lvoid_p)lsrc;
    asm volatile("global_store_async_from_lds_b128 %0, %1, off"
                 :: "v"(gdst), "v"(lo) : "memory");
#endif
}

__device__ static inline void wait_asynccnt0() {
#if __has_builtin(__builtin_amdgcn_s_wait_asynccnt)
    __builtin_amdgcn_s_wait_asynccnt(0);
#else
    asm volatile("s_wait_asynccnt 0" ::: "memory");
#endif
}

// ---------------- per-row LSP stabilization (4 passes, fully in VGPRs) ----
__device__ static inline void lsp_adjust(float (&reg)[32]) {
    const float lo = kMinD;
    const float hi = kPiF - kMinD;
#pragma unroll
    for (int it = 0; it < 4; ++it) {
        float c = reg[0];
#pragma unroll
        for (int i = 1; i < 32; ++i) {
            float nxt = reg[i];
            float s = 0.5f * fmaxf(kMinD - (nxt - c), 0.0f);
            reg[i - 1] = c - s;   // previous element, pushed down
            c = nxt + s;          // current element, pushed up, carried on
        }
        reg[31] = c;
#pragma unroll
        for (int i = 0; i < 32; ++i)
            reg[i] = fminf(fmaxf(reg[i], lo), hi);
    }
}

__global__ __launch_bounds__(ROWS_PER_BLOCK)
void lsp_stability_kernel(const float* __restrict__ w,
                          float* __restrict__ out,
                          int rows) {
    __shared__ float tile[TILE_WORDS];
    const int tid = threadIdx.x;
    const int row0 = blockIdx.x * ROWS_PER_BLOCK;

    if (row0 + ROWS_PER_BLOCK <= rows) {
        const float* gbase = w   + (size_t)row0 * NP1;  // 16B-aligned (33792B tiles)
        float*       obase = out + (size_t)row0 * NP1;

        // ---- Phase 1: async DMA the whole tile into LDS, fully coalesced ----
#pragma unroll
        for (int k = 0; k < 8; ++k) {
            int v4 = k * ROWS_PER_BLOCK + tid;          // float4 chunk index
            async_load_b128(gbase + (size_t)v4 * 4, &tile[v4 * 4]);
        }
        if (tid < (TILE_VEC4 - 8 * ROWS_PER_BLOCK)) {   // 64 remaining chunks
            int v4 = 8 * ROWS_PER_BLOCK + tid;
            async_load_b128(gbase + (size_t)v4 * 4, &tile[v4 * 4]);
        }
        wait_asynccnt0();
        __syncthreads();

        // ---- Phase 2: one row per thread; stride-33 (odd) => no LDS bank conflicts
        const int base = tid * NP1;
        float reg[32];
#pragma unroll
        for (int i = 0; i < 32; ++i) reg[i] = tile[base + 1 + i];

        lsp_adjust(reg);

#pragma unroll
        for (int i = 0; i < 32; ++i) tile[base + 1 + i] = reg[i];
        // tile[base + 0] (K) is untouched and streams straight back out.
        __syncthreads();

        // ---- Phase 3: async DMA LDS -> global, fully coalesced ----
#pragma unroll
        for (int k = 0; k < 8; ++k) {
            int v4 = k * ROWS_PER_BLOCK + tid;
            async_store_b128(obase + (size_t)v4 * 4, &tile[v4 * 4]);
        }
        if (tid < (TILE_VEC4 - 8 * ROWS_PER_BLOCK)) {
            int v4 = 8 * ROWS_PER_BLOCK + tid;
            async_store_b128(obase + (size_t)v4 * 4, &tile[v4 * 4]);
        }
        wait_asynccnt0();   // LDS must stay live until the DMA engine has read it
    } else {
        // Tail path (not taken for the 1,048,576-row shape): direct global access.
        int row = row0 + tid;
        if (row < rows) {
            const float* g = w   + (size_t)row * NP1;
            float*       o = out + (size_t)row * NP1;
            float K = g[0];
            float reg[32];
#pragma unroll
            for (int i = 0; i < 32; ++i) reg[i] = g[1 + i];
            lsp_adjust(reg);
            o[0] = K;
#pragma unroll
            for (int i = 0; i < 32; ++i) o[1 + i] = reg[i];
        }
    }
}

extern "C" void kernel_launch(void* const* d_in, const int* in_sizes, int n_in,
                              void* d_out, int out_size, void* d_ws, size_t ws_size,
                              hipStream_t stream) {
    (void)n_in; (void)out_size; (void)d_ws; (void)ws_size;
    const float* w   = (const float*)d_in[0];
    float*       out = (float*)d_out;

    const int total = in_sizes[0];          // 256*4096*33
    const int rows  = total / NP1;          // 1,048,576
    const int blocks = (rows + ROWS_PER_BLOCK - 1) / ROWS_PER_BLOCK;  // 4096

    lsp_stability_kernel<<<blocks, ROWS_PER_BLOCK, 0, stream>>>(w, out, rows);
}